// ImplicitModel_13357348291013
// MI455X (gfx1250) — compile-verified
//
#include <hip/hip_runtime.h>

typedef __attribute__((ext_vector_type(2))) float v2f;
typedef __attribute__((ext_vector_type(8))) float v8f;

#define N_DIM 1024
#define P_DIM 512
#define Q_DIM 512
#define M_DIM 2048
#define KAPPA 0.95f
#define TOL   3e-6f
#define MITR  300

// GEMM tiling: block = 128 x 128, 8 waves arranged 4x2, wave tile 32x64
// (2x4 accumulators of 16x16). Double-buffered LDS staging, BK=16.
#define BM  128
#define BN  128
#define BK  16
#define LDA 20    // As pitch: 16B-aligned rows, bank-spread (gcd(20,64)=4 -> 16 distinct)
#define LDX 132   // Xs pitch: 16B-aligned rows

// ---------------------------------------------------------------------------
// f32 WMMA: D(16x16,f32) = A(16x4,f32) * B(4x16,f32) + C
__device__ __forceinline__ v8f wmma4(v2f a, v2f b, v8f c) {
  return __builtin_amdgcn_wmma_f32_16x16x4_f32(false, a, false, b, (short)0, c,
                                               false, false);
}

// One K-tile of 16 (4 wmma steps of K=4), 2x4 accumulators per wave.
// A frag (16x4 f32): lane<16 -> rows M=lane, K=k,k+1 ; lane>=16 -> K=k+2,k+3
// B frag (4x16 f32): lane<16 -> cols N=lane, K=k,k+1 ; lane>=16 -> K=k+2,k+3
__device__ __forceinline__ void mma_ktile(const float* As, const float* Xs,
                                          int wr, int wc, int lane,
                                          v8f acc[2][4]) {
  const int lm = lane & 15, lh = lane >> 4;
#pragma unroll
  for (int kk = 0; kk < BK; kk += 4) {
    v2f a[2], b[4];
#pragma unroll
    for (int ti = 0; ti < 2; ++ti) {
      const float* p = As + (wr * 32 + ti * 16 + lm) * LDA + kk + lh * 2;
      a[ti].x = p[0];
      a[ti].y = p[1];
    }
#pragma unroll
    for (int tj = 0; tj < 4; ++tj) {
      const int cc = wc * 64 + tj * 16 + lm;
      const int kr = kk + lh * 2;
      b[tj].x = Xs[kr * LDX + cc];
      b[tj].y = Xs[(kr + 1) * LDX + cc];
    }
#pragma unroll
    for (int ti = 0; ti < 2; ++ti)
#pragma unroll
      for (int tj = 0; tj < 4; ++tj)
        acc[ti][tj] = wmma4(a[ti], b[tj], acc[ti][tj]);
  }
}

// ---------------------------------------------------------------------------
// Register staging for double-buffered LDS tiles (256 threads/block).
struct Frag4x2 { float4 v0, v1; };

// A-tile: 128 rows x 16 k (row-major source, leading dim ld) = 512 float4.
__device__ __forceinline__ Frag4x2 ldg_As(const float* A, int row0, int k0,
                                          int ld) {
  Frag4x2 f;
  {
    const int idx = threadIdx.x;
    const int r = idx >> 2, c = (idx & 3) << 2;
    f.v0 = *(const float4*)(A + (size_t)(row0 + r) * ld + k0 + c);
  }
  {
    const int idx = threadIdx.x + 256;
    const int r = idx >> 2, c = (idx & 3) << 2;
    f.v1 = *(const float4*)(A + (size_t)(row0 + r) * ld + k0 + c);
  }
  return f;
}
__device__ __forceinline__ void sts_As(float* As, const Frag4x2& f) {
  {
    const int idx = threadIdx.x;
    const int r = idx >> 2, c = (idx & 3) << 2;
    *(float4*)(As + r * LDA + c) = f.v0;
  }
  {
    const int idx = threadIdx.x + 256;
    const int r = idx >> 2, c = (idx & 3) << 2;
    *(float4*)(As + r * LDA + c) = f.v1;
  }
}

// B-tile: 16 k x 128 cols. NT=false: row-major [K][cols] source.
// NT=true: row-major [cols][K] source (B = U^T), transposed on LDS store.
template <bool NT>
__device__ __forceinline__ Frag4x2 ldg_Xs(const float* X, int k0, int col0,
                                          int ld) {
  Frag4x2 f;
#pragma unroll
  for (int u = 0; u < 2; ++u) {
    const int idx = threadIdx.x + u * 256;
    float4 v;
    if (NT) {
      const int j = idx >> 2, kq = (idx & 3) << 2;
      v = *(const float4*)(X + (size_t)(col0 + j) * ld + k0 + kq);
    } else {
      const int r = idx >> 5, c = (idx & 31) << 2;
      v = *(const float4*)(X + (size_t)(k0 + r) * ld + col0 + c);
    }
    if (u == 0) f.v0 = v; else f.v1 = v;
  }
  return f;
}
template <bool NT>
__device__ __forceinline__ void sts_Xs(float* Xs, const Frag4x2& f) {
#pragma unroll
  for (int u = 0; u < 2; ++u) {
    const int idx = threadIdx.x + u * 256;
    const float4 v = (u == 0) ? f.v0 : f.v1;
    if (NT) {
      const int j = idx >> 2, kq = (idx & 3) << 2;
      Xs[(kq + 0) * LDX + j] = v.x;
      Xs[(kq + 1) * LDX + j] = v.y;
      Xs[(kq + 2) * LDX + j] = v.z;
      Xs[(kq + 3) * LDX + j] = v.w;
    } else {
      const int r = idx >> 5, c = (idx & 31) << 2;
      *(float4*)(Xs + r * LDX + c) = v;
    }
  }
}

// Double-buffered pipelined K loop: global loads issue ahead of the WMMA
// block, LDS stores land after it; one barrier per K-tile.
template <bool NT>
__device__ __forceinline__ void gemm_pipeline(
    const float* __restrict__ Amat, int lda, const float* __restrict__ Bmat,
    int ldb, int Kdim, int row0, int col0, float (*AsB)[BM * LDA],
    float (*XsB)[BK * LDX], int wr, int wc, int lane, v8f acc[2][4]) {
  Frag4x2 fa = ldg_As(Amat, row0, 0, lda);
  Frag4x2 fx = ldg_Xs<NT>(Bmat, 0, col0, ldb);
  sts_As(AsB[0], fa);
  sts_Xs<NT>(XsB[0], fx);
  __syncthreads();
  int buf = 0;
  for (int k0 = 0; k0 < Kdim - BK; k0 += BK) {
    fa = ldg_As(Amat, row0, k0 + BK, lda);
    fx = ldg_Xs<NT>(Bmat, k0 + BK, col0, ldb);
    mma_ktile(AsB[buf], XsB[buf], wr, wc, lane, acc);
    sts_As(AsB[buf ^ 1], fa);
    sts_Xs<NT>(XsB[buf ^ 1], fx);
    __syncthreads();
    buf ^= 1;
  }
  mma_ktile(AsB[buf], XsB[buf], wr, wc, lane, acc);
  __syncthreads();  // LDS safe for reuse by caller
}

// ---------------------------------------------------------------------------
// BU = B @ U^T   (B: N x P row-major, U: M x P row-major, BU: N x M row-major)
__global__ __launch_bounds__(256) void bu_gemm(const float* __restrict__ B,
                                               const float* __restrict__ U,
                                               float* __restrict__ BU) {
  __shared__ alignas(16) float As[2][BM * LDA];
  __shared__ alignas(16) float Xs[2][BK * LDX];
  const int row0 = blockIdx.y * BM;
  const int col0 = blockIdx.x * BN;
  const int lane = threadIdx.x & 31, wave = threadIdx.x >> 5;
  const int wr = wave >> 1, wc = wave & 1;
  v8f acc[2][4] = {};
  gemm_pipeline<true>(B, P_DIM, U, P_DIM, P_DIM, row0, col0, As, Xs, wr, wc,
                      lane, acc);
  const int lm = lane & 15, lh = lane >> 4;
#pragma unroll
  for (int ti = 0; ti < 2; ++ti)
#pragma unroll
    for (int tj = 0; tj < 4; ++tj) {
      const int gc = col0 + wc * 64 + tj * 16 + lm;
#pragma unroll
      for (int r = 0; r < 8; ++r) {
        const int gr = row0 + wr * 32 + ti * 16 + lh * 8 + r;
        BU[(size_t)gr * M_DIM + gc] = acc[ti][tj][r];
      }
    }
}

// ---------------------------------------------------------------------------
// One Picard iteration: Xout = relu(Ap @ Xin + BU); errbits = max|Xout - Xin|.
// If converged, degrade to a pass-through tile copy (keeps ping-pong parity
// deterministic under graph replay; branch is block-uniform, so EXEC stays
// all-ones around the WMMA chain).
__global__ __launch_bounds__(256) void iter_gemm(const float* __restrict__ Ap,
                                                 const float* __restrict__ Xin,
                                                 const float* __restrict__ BU,
                                                 float* __restrict__ Xout,
                                                 unsigned* __restrict__ errbits,
                                                 const int* __restrict__ flag) {
  __shared__ alignas(16) float As[2][BM * LDA];
  __shared__ alignas(16) float Xs[2][BK * LDX];
  const int row0 = blockIdx.y * BM;
  const int col0 = blockIdx.x * BN;

  if (*flag) {  // uniform across the block
#pragma unroll
    for (int u = 0; u < 16; ++u) {
      const int idx = threadIdx.x + u * 256;
      const int r = idx >> 5, c = (idx & 31) << 2;
      const float4 v =
          *(const float4*)(Xin + (size_t)(row0 + r) * M_DIM + col0 + c);
      *(float4*)(Xout + (size_t)(row0 + r) * M_DIM + col0 + c) = v;
    }
    return;
  }

  const int lane = threadIdx.x & 31, wave = threadIdx.x >> 5;
  const int wr = wave >> 1, wc = wave & 1;
  const int lm = lane & 15, lh = lane >> 4;

  // Preload BU into the accumulators (C operand of the WMMA chain).
  v8f acc[2][4];
#pragma unroll
  for (int ti = 0; ti < 2; ++ti)
#pragma unroll
    for (int tj = 0; tj < 4; ++tj) {
      const int gc = col0 + wc * 64 + tj * 16 + lm;
#pragma unroll
      for (int r = 0; r < 8; ++r) {
        const int gr = row0 + wr * 32 + ti * 16 + lh * 8 + r;
        acc[ti][tj][r] = BU[(size_t)gr * M_DIM + gc];
      }
    }

  gemm_pipeline<false>(Ap, N_DIM, Xin, M_DIM, N_DIM, row0, col0, As, Xs, wr,
                       wc, lane, acc);

  float lmax = 0.f;
#pragma unroll
  for (int ti = 0; ti < 2; ++ti)
#pragma unroll
    for (int tj = 0; tj < 4; ++tj) {
      const int gc = col0 + wc * 64 + tj * 16 + lm;
#pragma unroll
      for (int r = 0; r < 8; ++r) {
        const int gr = row0 + wr * 32 + ti * 16 + lh * 8 + r;
        float y = acc[ti][tj][r];
        y = y > 0.f ? y : 0.f;  // relu
        const float xo = Xin[(size_t)gr * M_DIM + gc];
        lmax = fmaxf(lmax, fabsf(y - xo));
        Xout[(size_t)gr * M_DIM + gc] = y;
      }
    }

  // block-wide max reduce (reuse As as scratch; pipeline ended with a barrier)
  float* red = &As[0][0];
  red[threadIdx.x] = lmax;
  __syncthreads();
  for (int s = 128; s > 0; s >>= 1) {
    if ((int)threadIdx.x < s)
      red[threadIdx.x] = fmaxf(red[threadIdx.x], red[threadIdx.x + s]);
    __syncthreads();
  }
  if (threadIdx.x == 0)
    atomicMax(errbits, __float_as_uint(red[0]));  // valid: values >= 0
}

__global__ void check_converged(unsigned* errbits, int* flag) {
  if (!*flag) {
    if (__uint_as_float(*errbits) < TOL) *flag = 1;
  }
  *errbits = 0u;
}

// ---------------------------------------------------------------------------
// Out[m][q] = sum_n C[q][n] X[n][m] + sum_p D[q][p] U[m][p]  (transposed store)
__global__ __launch_bounds__(256) void out_gemm(const float* __restrict__ C,
                                                const float* __restrict__ D,
                                                const float* __restrict__ X,
                                                const float* __restrict__ U,
                                                float* __restrict__ Out) {
  __shared__ alignas(16) float As[2][BM * LDA];
  __shared__ alignas(16) float Xs[2][BK * LDX];
  const int row0 = blockIdx.y * BM;  // q
  const int col0 = blockIdx.x * BN;  // m
  const int lane = threadIdx.x & 31, wave = threadIdx.x >> 5;
  const int wr = wave >> 1, wc = wave & 1;
  v8f acc[2][4] = {};
  gemm_pipeline<false>(C, N_DIM, X, M_DIM, N_DIM, row0, col0, As, Xs, wr, wc,
                       lane, acc);  // C @ X   (NN)
  gemm_pipeline<true>(D, P_DIM, U, P_DIM, P_DIM, row0, col0, As, Xs, wr, wc,
                      lane, acc);   // D @ U^T (NT)
  const int lm = lane & 15, lh = lane >> 4;
#pragma unroll
  for (int ti = 0; ti < 2; ++ti)
#pragma unroll
    for (int tj = 0; tj < 4; ++tj) {
      const int gm = col0 + wc * 64 + tj * 16 + lm;
#pragma unroll
      for (int r = 0; r < 8; ++r) {
        const int gq = row0 + wr * 32 + ti * 16 + lh * 8 + r;
        Out[(size_t)gm * Q_DIM + gq] = acc[ti][tj][r];
      }
    }
}

// ---------------------------------------------------------------------------
// Row-wise projection of A onto the L1 ball of radius KAPPA.
// One block per row: bitonic sort (desc) of |row| in LDS, blocked scan, rho/theta.
__global__ __launch_bounds__(256) void project_rows(const float* __restrict__ A,
                                                    float* __restrict__ Ap) {
  __shared__ float s[N_DIM];
  __shared__ float c[N_DIM];
  __shared__ float red[256];
  __shared__ float sh_theta;
  __shared__ int sh_need;
  const int row = blockIdx.x;
  const int t = threadIdx.x;
  const float* a = A + (size_t)row * N_DIM;

  for (int i = t; i < N_DIM; i += 256) s[i] = fabsf(a[i]);
  __syncthreads();

  // bitonic sort, descending
  for (int k = 2; k <= N_DIM; k <<= 1) {
    for (int j = k >> 1; j > 0; j >>= 1) {
      for (int p = t; p < N_DIM / 2; p += 256) {
        const int i = ((p & ~(j - 1)) << 1) | (p & (j - 1));
        const int ixj = i + j;
        const bool desc = ((i & k) == 0);
        const float x = s[i], y = s[ixj];
        if (desc ? (x < y) : (x > y)) { s[i] = y; s[ixj] = x; }
      }
      __syncthreads();
    }
  }

  // blocked inclusive scan: thread t owns elements [4t, 4t+4)
  const int base = t * 4;
  const float v0 = s[base], v1 = s[base + 1], v2 = s[base + 2], v3 = s[base + 3];
  const float i0 = v0, i1 = i0 + v1, i2 = i1 + v2, i3 = i2 + v3;
  red[t] = i3;
  __syncthreads();
  for (int off = 1; off < 256; off <<= 1) {
    const float add = (t >= off) ? red[t - off] : 0.f;
    __syncthreads();
    red[t] += add;
    __syncthreads();
  }
  const float offset = (t > 0) ? red[t - 1] : 0.f;
  c[base] = i0 + offset; c[base + 1] = i1 + offset;
  c[base + 2] = i2 + offset; c[base + 3] = i3 + offset;
  __syncthreads();

  // rho = #{k : srt_k - (css_k - v)/k > 0}  <=>  srt_k * k > css_k - v
  int cnt = 0;
#pragma unroll
  for (int u = 0; u < 4; ++u) {
    const int idx = base + u;
    if (s[idx] * (float)(idx + 1) > c[idx] - KAPPA) ++cnt;
  }
  red[t] = (float)cnt;
  __syncthreads();
  for (int sr = 128; sr > 0; sr >>= 1) {
    if (t < sr) red[t] += red[t + sr];
    __syncthreads();
  }
  if (t == 0) {
    const int rho = (int)red[0];  // >= 1 always
    sh_theta = (c[rho - 1] - KAPPA) / (float)rho;
    sh_need = (c[N_DIM - 1] > KAPPA) ? 1 : 0;
  }
  __syncthreads();
  const float theta = sh_theta;
  const int need = sh_need;
  for (int i = t; i < N_DIM; i += 256) {
    const float av = a[i];
    const float pr = copysignf(fmaxf(fabsf(av) - theta, 0.f), av);
    Ap[(size_t)row * N_DIM + i] = need ? pr : av;
  }
}

__global__ void init_state(float* __restrict__ X0, unsigned* errbits,
                           int* flag) {
  const size_t i = (size_t)blockIdx.x * blockDim.x + threadIdx.x;
  X0[i] = 0.f;
  if (i == 0) { *errbits = 0u; *flag = 0; }
}

// ---------------------------------------------------------------------------
extern "C" void kernel_launch(void* const* d_in, const int* in_sizes, int n_in,
                              void* d_out, int out_size, void* d_ws,
                              size_t ws_size, hipStream_t stream) {
  (void)in_sizes; (void)n_in; (void)out_size; (void)ws_size;
  const float* U = (const float*)d_in[0];  // M x P
  const float* A = (const float*)d_in[1];  // N x N
  const float* B = (const float*)d_in[2];  // N x P
  const float* C = (const float*)d_in[3];  // Q x N
  const float* D = (const float*)d_in[4];  // Q x P

  float* ws = (float*)d_ws;
  float* Ap = ws;                                   // N*N
  float* BU = Ap + (size_t)N_DIM * N_DIM;           // N*M
  float* X0 = BU + (size_t)N_DIM * M_DIM;           // N*M
  float* X1 = X0 + (size_t)N_DIM * M_DIM;           // N*M
  unsigned* errbits = (unsigned*)(X1 + (size_t)N_DIM * M_DIM);
  int* flag = (int*)(errbits + 1);
  float* Out = (float*)d_out;                       // M x Q

  init_state<<<(N_DIM * M_DIM) / 256, 256, 0, stream>>>(X0, errbits, flag);
  project_rows<<<N_DIM, 256, 0, stream>>>(A, Ap);

  dim3 g1(M_DIM / BN, N_DIM / BM);
  bu_gemm<<<g1, 256, 0, stream>>>(B, U, BU);

  float* bufs[2] = {X0, X1};
  for (int it = 0; it < MITR; ++it) {
    iter_gemm<<<g1, 256, 0, stream>>>(Ap, bufs[it & 1], BU, bufs[(it + 1) & 1],
                                      errbits, flag);
    check_converged<<<1, 1, 0, stream>>>(errbits, flag);
  }

  dim3 g2(M_DIM / BN, Q_DIM / BM);
  out_gemm<<<g2, 256, 0, stream>>>(C, D, bufs[MITR & 1], U, Out);
}